// BilateralFilter_61168924230190
// MI455X (gfx1250) — compile-verified
//
#include <hip/hip_runtime.h>

namespace {
constexpr int KS   = 7;
constexpr int PADR = 3;
constexpr int IMH  = 512;
constexpr int IMW  = 512;
constexpr int NB   = 4;

constexpr int TW   = 64;               // tile width  (output)
constexpr int TH   = 16;               // tile height (output)
constexpr int HLW  = TW + 2 * PADR;    // 70 halo width
constexpr int HLH  = TH + 2 * PADR;    // 22 halo height
constexpr int LSTR = 72;               // padded LDS row stride (floats)
constexpr int NTHR = 256;              // 8 waves (wave32)
constexpr int PPT  = (TW * TH) / NTHR; // 4 pixels per thread
}

// CDNA5 async global->LDS path (guarded: falls back to ds_store staging if
// this toolchain doesn't expose the gfx1250 async builtins).
#if defined(__HIP_DEVICE_COMPILE__) && defined(__gfx1250__) && \
    __has_builtin(__builtin_amdgcn_global_load_async_to_lds_b32) && \
    __has_builtin(__builtin_amdgcn_s_wait_asynccnt)
#define USE_ASYNC_LDS 1
#else
#define USE_ASYNC_LDS 0
#endif

typedef __attribute__((address_space(1))) int* as1_i32p;
typedef __attribute__((address_space(3))) int* as3_i32p;

__device__ __forceinline__ int reflect_idx(int i, int n) {
  i = (i < 0) ? -i : i;            // reflect (no edge repeat), PAD < n so one pass suffices
  i = (i >= n) ? (2 * n - 2 - i) : i;
  return i;
}

__global__ __launch_bounds__(NTHR)
void bilateral7x7_kernel(const float* __restrict__ x,
                         const float* __restrict__ psx,
                         const float* __restrict__ psy,
                         const float* __restrict__ psr,
                         float* __restrict__ out)
{
  __shared__ float tile[HLH * LSTR];
  __shared__ float slog[KS * KS];

  const int tid = threadIdx.x;
  constexpr int TPX = IMW / TW;   // 8
  constexpr int TPY = IMH / TH;   // 32
  int tb = blockIdx.x;
  const int b = tb / (TPX * TPY);
  tb -= b * (TPX * TPY);
  const int ty = tb / TPX;
  const int tx = tb - ty * TPX;
  const int gx0 = tx * TW;
  const int gy0 = ty * TH;
  const float* img = x + (size_t)b * IMH * IMW;

  // ---- stage reflect-padded halo tile into LDS ----
  for (int idx = tid; idx < HLH * HLW; idx += NTHR) {
    const int ly = idx / HLW;
    const int lx = idx - ly * HLW;
    const int gy = reflect_idx(gy0 + ly - PADR, IMH);
    const int gx = reflect_idx(gx0 + lx - PADR, IMW);
#if USE_ASYNC_LDS
    __builtin_amdgcn_global_load_async_to_lds_b32(
        (as1_i32p)(img + gy * IMW + gx),
        (as3_i32p)&tile[ly * LSTR + lx],
        /*imm offset*/ 0, /*cpol*/ 0);
#else
    tile[ly * LSTR + lx] = img[gy * IMW + gx];
#endif
  }

  // ---- spatial weights in log2 domain, once per block ----
  const float LOG2E = 1.4426950408889634f;
  const float sx = psx[0], sy = psy[0], sr = psr[0];
  const float ax = LOG2E / (2.0f * sx * sx);          // columns (jj) <-> sigma_sx
  const float ay = LOG2E / (2.0f * sy * sy);          // rows    (ii) <-> sigma_sy
  const float cr = LOG2E / (2.0f * sr * sr + 1e-8f);  // range coefficient
  if (tid < KS * KS) {
    const int i = tid / KS;
    const int j = tid - i * KS;
    const float di = (float)(i - PADR);
    const float dj = (float)(j - PADR);
    slog[tid] = dj * dj * ax + di * di * ay;          // positive log2-exponent
  }

#if USE_ASYNC_LDS
  __builtin_amdgcn_s_wait_asynccnt(0);
#endif
  __syncthreads();

  float sl[KS * KS];
#pragma unroll
  for (int k = 0; k < KS * KS; ++k) sl[k] = slog[k];  // broadcast reads -> regs

  const int lx  = tid & (TW - 1);
  const int lyb = tid >> 6;                           // 0..3
  float* obase = out + (size_t)b * IMH * IMW;

#pragma unroll
  for (int pp = 0; pp < PPT; ++pp) {
    const int ly = lyb + pp * (NTHR / TW);            // rows 0..15, stride 4
    const float* trow = &tile[ly * LSTR + lx];        // 7x7 window origin
    const float c = trow[PADR * LSTR + PADR];
    float num = 0.0f, den = 0.0f;
#pragma unroll
    for (int i = 0; i < KS; ++i) {
#pragma unroll
      for (int j = 0; j < KS; ++j) {
        const float p = trow[i * LSTR + j];
        const float d = c - p;
        // kern = spatial * range = exp2(-(sl + d^2 * cr)) : ONE v_exp_f32 per tap
        const float t = __builtin_fmaf(d * d, -cr, -sl[i * KS + j]);
        const float e = __builtin_amdgcn_exp2f(t);
        den += e;
        num = __builtin_fmaf(p, e, num);
      }
    }
    obase[(gy0 + ly) * IMW + gx0 + lx] =
        num * __builtin_amdgcn_rcpf(den + 1e-8f);
  }
}

extern "C" void kernel_launch(void* const* d_in, const int* in_sizes, int n_in,
                              void* d_out, int out_size, void* d_ws, size_t ws_size,
                              hipStream_t stream) {
  const float* x  = (const float*)d_in[0];
  const float* sx = (const float*)d_in[1];
  const float* sy = (const float*)d_in[2];
  const float* sr = (const float*)d_in[3];
  float* out = (float*)d_out;
  const int nblocks = NB * (IMH / TH) * (IMW / TW);   // 1024
  hipLaunchKernelGGL(bilateral7x7_kernel, dim3(nblocks), dim3(NTHR), 0, stream,
                     x, sx, sy, sr, out);
}